// Projection_30777735643395
// MI455X (gfx1250) — compile-verified
//
#include <hip/hip_runtime.h>

// Trilinear multi-resolution feature sampling, MI455X (gfx1250, wave32).
// One wave32 per sample point (grid.y = batch, no integer division):
//   level1 (64^3 x 32):  lane c -> 1 channel,  8x global_load_b32
//   level2 (32^3 x 64):  lane c -> 2 channels, 8x global_load_b64
//   level3 (16^3 x128):  lane c -> 4 channels, 8x global_load_b128
// Working set (84 MB of grids) is global-L2 resident; kernel is an
// L2-gather-bandwidth problem, so the levers are full-width coalesced VMEM,
// global_prefetch_b8 for the wide levels, and high wave32 occupancy.

#define WAVES_PER_BLOCK 8

struct CornerSet {
  const float* p[8];
  float wx, wx2, wy, wy2, wz, wz2;
};

__device__ __forceinline__ void axis_setup(float c, float scale, float hi,
                                           int& i1, int& i2, float& w, float& w2) {
  float ic = fminf(fmaxf(c * scale, 0.01f), hi);  // jnp.clip(c*scale, 0.01, hi)
  float f1 = floorf(ic);
  float f2 = ceilf(ic);
  i1 = (int)f1;
  i2 = (int)f2;
  w  = ic - f1;   // wx
  w2 = f2 - ic;   // wx2
}

template <int S, int C, int V>
__device__ __forceinline__ CornerSet corner_setup(const float* __restrict__ featb,
                                                  float cx, float cy, float cz,
                                                  int lane) {
  const float scale = (float)S;            // 0.5^p * 128 == S for these levels
  const float hi    = (float)(S - 1.01);   // match float64->float32 rounding
  int x1, x2, y1, y2, z1, z2;
  CornerSet K;
  axis_setup(cx, scale, hi, x1, x2, K.wx, K.wx2);
  axis_setup(cy, scale, hi, y1, y2, K.wy, K.wy2);
  axis_setup(cz, scale, hi, z1, z2, K.wz, K.wz2);
  const float* base = featb + lane * V;    // featb already offset to batch b
  auto at = [&](int x, int y, int z) {
    return base + (size_t)(((x * S + y) * S + z) * C);
  };
  // order: (x2,y2,z2)(x1,y2,z2)(x2,y1,z2)(x1,y1,z2)(x2,y2,z1)(x1,y2,z1)(x2,y1,z1)(x1,y1,z1)
  K.p[0] = at(x2, y2, z2); K.p[1] = at(x1, y2, z2);
  K.p[2] = at(x2, y1, z2); K.p[3] = at(x1, y1, z2);
  K.p[4] = at(x2, y2, z1); K.p[5] = at(x1, y2, z1);
  K.p[6] = at(x2, y1, z1); K.p[7] = at(x1, y1, z1);
  return K;
}

template <int V>
__device__ __forceinline__ void loadV(const float* p, float d[V]) {
  if constexpr (V == 4) {
    float4 t = *(const float4*)p;          // global_load_b128
    d[0] = t.x; d[1] = t.y; d[2] = t.z; d[3] = t.w;
  } else if constexpr (V == 2) {
    float2 t = *(const float2*)p;          // global_load_b64
    d[0] = t.x; d[1] = t.y;
  } else {
    d[0] = *p;                             // global_load_b32
  }
}

template <int V>
__device__ __forceinline__ void trilerp_finish(const CornerSet& K, float out[V]) {
  float g[8][V];
#pragma unroll
  for (int i = 0; i < 8; ++i) loadV<V>(K.p[i], g[i]);  // all 8 gathers in flight
#pragma unroll
  for (int v = 0; v < V; ++v) {
    // z = z2 plane: lerp_x at y2/y1, then lerp_y (matches reference ordering)
    float ly2 = (g[0][v] * K.wx + g[1][v] * K.wx2) * K.wy +
                (g[2][v] * K.wx + g[3][v] * K.wx2) * K.wy2;
    // z = z1 plane
    float ly1 = (g[4][v] * K.wx + g[5][v] * K.wx2) * K.wy +
                (g[6][v] * K.wx + g[7][v] * K.wx2) * K.wy2;
    out[v] = ly2 * K.wz + ly1 * K.wz2;
  }
}

__global__ __launch_bounds__(WAVES_PER_BLOCK * 32)
void trilerp227_kernel(const float* __restrict__ f1, const float* __restrict__ f2,
                       const float* __restrict__ f3, const float* __restrict__ coords,
                       const float* __restrict__ mesh, float* __restrict__ out,
                       int N) {
  const int n = blockIdx.x * WAVES_PER_BLOCK + (threadIdx.x >> 5);  // point in batch
  if (n >= N) return;                     // wave-uniform exit
  const int lane = threadIdx.x & 31;
  const int b = blockIdx.y;               // batch -> no integer division

  // Per-batch base pointers (SALU-friendly: b is wave/grid uniform).
  const float* f1b = f1 + (size_t)b * (64 * 64 * 64 * 32);
  const float* f2b = f2 + (size_t)b * (32 * 32 * 32 * 64);
  const float* f3b = f3 + (size_t)b * (16 * 16 * 16 * 128);
  const size_t row = (size_t)b * N + n;   // flat point id

  const float* cp = coords + row * 3;     // one global_load_b96 per wave
  const float cx = cp[0], cy = cp[1], cz = cp[2];

  CornerSet k1 = corner_setup<64,  32, 1>(f1b, cx, cy, cz, lane);
  CornerSet k2 = corner_setup<32,  64, 2>(f2b, cx, cy, cz, lane);
  CornerSet k3 = corner_setup<16, 128, 4>(f3b, cx, cy, cz, lane);

  // gfx1250 global_prefetch_b8: warm the wide-level corner lines while the
  // level-1 gather + lerp runs.
#pragma unroll
  for (int i = 0; i < 8; ++i) {
    __builtin_prefetch((const void*)k3.p[i], 0, 3);
    __builtin_prefetch((const void*)k2.p[i], 0, 3);
  }

  float r1[1], r2[2], r3[4];
  trilerp_finish<1>(k1, r1);
  trilerp_finish<2>(k2, r2);
  trilerp_finish<4>(k3, r3);

  // Output row: [lvl1(32) | lvl2(64) | lvl3(128) | mesh(3)] = 227 floats.
  // 227 is odd -> rows only 4B aligned, so use lane-consecutive b32 stores
  // (fully coalesced into lines by the memory system).
  const size_t o = row * 227;
  out[o + lane] = r1[0];
  out[o + 32 + 2 * lane + 0] = r2[0];
  out[o + 32 + 2 * lane + 1] = r2[1];
#pragma unroll
  for (int v = 0; v < 4; ++v) out[o + 96 + 4 * lane + v] = r3[v];
  if (lane < 3) out[o + 224 + lane] = mesh[row * 3 + lane];
}

extern "C" void kernel_launch(void* const* d_in, const int* in_sizes, int n_in,
                              void* d_out, int out_size, void* d_ws, size_t ws_size,
                              hipStream_t stream) {
  // setup_inputs order: features0..features4, mesh_coords, mesh_features
  const float* f1     = (const float*)d_in[1];  // [B,64,64,64,32]
  const float* f2     = (const float*)d_in[2];  // [B,32,32,32,64]
  const float* f3     = (const float*)d_in[3];  // [B,16,16,16,128]
  const float* coords = (const float*)d_in[5];  // [B,N,3]
  const float* mesh   = (const float*)d_in[6];  // [B,N,3]
  float* out = (float*)d_out;                   // [B,N,227]

  int B = in_sizes[1] / (64 * 64 * 64 * 32);
  if (B < 1) B = 1;
  int N = (in_sizes[5] / 3) / B;

  dim3 grid((N + WAVES_PER_BLOCK - 1) / WAVES_PER_BLOCK, B, 1);
  trilerp227_kernel<<<grid, WAVES_PER_BLOCK * 32, 0, stream>>>(
      f1, f2, f3, coords, mesh, out, N);
}